// Pitch_84808424227064
// MI455X (gfx1250) — compile-verified
//
#include <hip/hip_runtime.h>
#include <math.h>

// ---- problem constants (derived from the reference) ----
#define T_IN      960000
#define PAD       1024
#define WPAD_LEN  (T_IN + 2*PAD)      // 962048
#define FRAME     512
#define NFRAMES   1879                // 962048 / 512
#define LAGS      189                 // ceil(16000/85)
#define WP_STRIDE 962560              // NFRAMES*512 + 512 slack (covers base+768 reads)
#define BATCH     8
#define XLEN      768                 // window: 512 + 16 (K pad) + 240 (16*15 B shift)
#define XA_LOG    544                 // zero-padded A source: 16 zeros + 512 + 16 zeros
#define XA_PAD    (XA_LOG + XA_LOG/4) // 680
#define XB_PAD    (XLEN + XLEN/4)     // 960
#define OUT_FRAMES 1876               // 960000//512 + 1
#define MED_LEN   1864                // 14 + 1879 - 30 + 1
#define WIN       30

typedef __attribute__((ext_vector_type(2))) float v2f;
typedef __attribute__((ext_vector_type(8))) float v8f;

// pad +1 float every 4: padded address advances by a constant 5 per K-step of 4
__device__ __forceinline__ int pidx4(int j) { return j + (j >> 2); }

// ---------------- kernel 1: reflect-pad + zero-pad into workspace ----------------
__global__ void build_wp(const float* __restrict__ w, float* __restrict__ wp) {
  long g = (long)blockIdx.x * blockDim.x + threadIdx.x;
  const long total = (long)BATCH * WP_STRIDE;
  if (g >= total) return;
  int b = (int)(g / WP_STRIDE);
  int t = (int)(g % WP_STRIDE);
  const float* wb = w + (long)b * T_IN;
  float v;
  if (t < PAD)             v = wb[PAD - t];                       // left reflect (edge excluded)
  else if (t < PAD + T_IN) v = wb[t - PAD];                       // body
  else if (t < WPAD_LEN)   v = wb[T_IN - 2 - (t - (PAD + T_IN))]; // right reflect
  else                     v = 0.0f;                              // zero pad (lags + slack)
  wp[g] = v;
}

// ---------------- kernel 2: per-frame NCCF via fp32 WMMA + argmax -> lag idx ----------------
// One wave per frame; 4 waves per block. D[m,n] = num(l = m + 16n):
//   A[m,k] = X(k-m)  (X zero-padded outside [0,512) -> xa array, no masking)
//   B[k,n] = x(k+16n) (xb array), K = 528 -> 132 x V_WMMA_F32_16X16X4_F32.
__global__ __launch_bounds__(128) void nccf_idx(const float* __restrict__ wp,
                                                int* __restrict__ idxbuf) {
  __shared__ float xa_all[4][XA_PAD];   // zero-padded A source
  __shared__ float xb_all[4][XB_PAD];   // raw window (B source)
  __shared__ float q_all[4][XLEN];      // inclusive prefix of squares
  __shared__ float ss_all[4][192];      // 1 / (eps + ||seg_l||)^2

  const int wave = threadIdx.x >> 5;
  const int lane = threadIdx.x & 31;
  const int b    = blockIdx.y;
  int f = blockIdx.x * 4 + wave;
  const bool valid = (f < NFRAMES);
  if (f > NFRAMES - 1) f = NFRAMES - 1;   // clamp: keep barriers uniform / EXEC full

  float* xa = xa_all[wave];
  float* xb = xb_all[wave];
  float* q  = q_all[wave];
  float* ss = ss_all[wave];

  const float* src = wp + (long)b * WP_STRIDE + (long)f * FRAME;

  // zero the A-source halo
  if (lane < 16) {
    xa[pidx4(lane)]       = 0.0f;
    xa[pidx4(528 + lane)] = 0.0f;
  }
  // load window (coalesced); mirror first 512 into xa at logical offset +16
  #pragma unroll
  for (int i = 0; i < XLEN / 32; ++i) {
    int j = i * 32 + lane;
    float v = src[j];
    xb[pidx4(j)] = v;
    if (j < FRAME) xa[pidx4(16 + j)] = v;
  }
  __syncthreads();

  // prefix sum of squares: 24-element chunk per lane + shfl_up scan of chunk totals
  {
    const int cs = XLEN / 32;   // 24
    const int j0 = lane * cs;
    float run = 0.0f;
    #pragma unroll
    for (int i = 0; i < cs; ++i) {
      float x = xb[pidx4(j0 + i)];
      run += x * x;
      q[j0 + i] = run;
    }
    float tot = run;
    #pragma unroll
    for (int d = 1; d < 32; d <<= 1) {
      float t = __shfl_up(tot, d, 32);
      if (lane >= d) tot += t;
    }
    const float off = tot - run;   // exclusive offset
    #pragma unroll
    for (int i = 0; i < cs; ++i) q[j0 + i] += off;
  }
  __syncthreads();

  // 1/s2n(l) for l in [0,192):  s2n = (eps + sqrt(sumsq(l)))^2 ; sumsq(l) = Q[l+511]-Q[l-1]
  #pragma unroll
  for (int i = 0; i < 6; ++i) {
    int l = i * 32 + lane;
    float sum = q[l + FRAME - 1] - (l ? q[l - 1] : 0.0f);
    sum = fmaxf(sum, 0.0f);
    float r = 1e-9f + sqrtf(sum);
    ss[l] = 1.0f / (r * r);
  }
  __syncthreads();

  // ---- WMMA accumulation (fp32, exact), branchless, strength-reduced addresses ----
  const int m    = lane & 15;          // M row (A) / N col (B)
  const int koff = (lane >> 4) * 2;    // K half per ISA 16x4 / 4x16 fp32 layout
  // initial padded addresses; each advances by 5 per K-step of 4 (20 per loop iteration)
  int pa0 = pidx4(16 + koff - m);
  int pa1 = pidx4(16 + koff - m + 1);
  int pb0 = pidx4(koff + 16 * m);
  int pb1 = pb0 + 1;                   // uniform +1 in padded space (jb%4 in {0,2})

  v8f acc = {};
  for (int c = 0; c < 132; c += 4) {
    // batch all LDS loads for 4 K-steps, then 4 WMMAs (lets dscnt waits overlap)
    float a00 = xa[pa0],      a01 = xa[pa1];
    float b00 = xb[pb0],      b01 = xb[pb1];
    float a10 = xa[pa0 + 5],  a11 = xa[pa1 + 5];
    float b10 = xb[pb0 + 5],  b11 = xb[pb1 + 5];
    float a20 = xa[pa0 + 10], a21 = xa[pa1 + 10];
    float b20 = xb[pb0 + 10], b21 = xb[pb1 + 10];
    float a30 = xa[pa0 + 15], a31 = xa[pa1 + 15];
    float b30 = xb[pb0 + 15], b31 = xb[pb1 + 15];
    v2f A0 = {a00, a01}, B0 = {b00, b01};
    v2f A1 = {a10, a11}, B1 = {b10, b11};
    v2f A2 = {a20, a21}, B2 = {b20, b21};
    v2f A3 = {a30, a31}, B3 = {b30, b31};
    acc = __builtin_amdgcn_wmma_f32_16x16x4_f32(false, A0, false, B0, (short)0, acc, false, false);
    acc = __builtin_amdgcn_wmma_f32_16x16x4_f32(false, A1, false, B1, (short)0, acc, false, false);
    acc = __builtin_amdgcn_wmma_f32_16x16x4_f32(false, A2, false, B2, (short)0, acc, false, false);
    acc = __builtin_amdgcn_wmma_f32_16x16x4_f32(false, A3, false, B3, (short)0, acc, false, false);
    pa0 += 20; pa1 += 20; pb0 += 20; pb1 += 20;
  }

  // ---- per-lane candidates; C/D layout: VGPR r -> M = r + 8*hi, N = lane&15 ----
  const int hi8   = (lane >> 4) * 8;
  const int nbase = 16 * (lane & 15);
  float bv = -3.4e38f; int bl = 1 << 20;   // best over l in [6,189]
  float hv = -3.4e38f; int hl = 1 << 20;   // best over l in [6,94]
  #pragma unroll
  for (int r = 0; r < 8; ++r) {
    int l = r + hi8 + nbase;
    if (l >= 6 && l <= LAGS) {
      float v = acc[r] * ss[l];            // s1n common scale: rank-invariant, dropped
      if (v > bv || (v == bv && l < bl)) { bv = v; bl = l; }
      if (l <= 94) {
        if (v > hv || (v == hv && l < hl)) { hv = v; hl = l; }
      }
    }
  }
  // wave32 butterfly argmax, first-index tie-break (matches jnp.argmax)
  #pragma unroll
  for (int d = 16; d >= 1; d >>= 1) {
    float v2 = __shfl_xor(bv, d, 32); int l2 = __shfl_xor(bl, d, 32);
    if (v2 > bv || (v2 == bv && l2 < bl)) { bv = v2; bl = l2; }
    float v3 = __shfl_xor(hv, d, 32); int l3 = __shfl_xor(hl, d, 32);
    if (v3 > hv || (v3 == hv && l3 < hl)) { hv = v3; hl = l3; }
  }
  if (valid && lane == 0) {
    int sel = (hv > 0.99f * bv) ? hl : bl;   // octave check
    idxbuf[b * NFRAMES + f] = sel;
  }
}

// ---------------- kernel 3: 30-tap median (15th smallest) + pitch ----------------
__global__ void median_out(const int* __restrict__ idxbuf, float* __restrict__ out) {
  int g = blockIdx.x * blockDim.x + threadIdx.x;
  if (g >= BATCH * OUT_FRAMES) return;
  int b = g / OUT_FRAMES;
  int o = g % OUT_FRAMES;
  if (o >= MED_LEN) { out[g] = 0.0f; return; }
  const int* ib = idxbuf + b * NFRAMES;
  int a[WIN];
  #pragma unroll
  for (int j = 0; j < WIN; ++j) {
    int s = o + j - 14;                 // left edge replicated 14x
    a[j] = ib[s < 0 ? 0 : s];
  }
  // stable-rank counting selection: rank == 14 -> 15th smallest
  int med = 0;
  #pragma unroll
  for (int i = 0; i < WIN; ++i) {
    int rank = 0;
    #pragma unroll
    for (int j = 0; j < WIN; ++j)
      rank += (a[j] < a[i] || (a[j] == a[i] && j < i)) ? 1 : 0;
    if (rank == 14) med = a[i];
  }
  out[g] = 16000.0f / (1e-9f + (float)med);
}

extern "C" void kernel_launch(void* const* d_in, const int* in_sizes, int n_in,
                              void* d_out, int out_size, void* d_ws, size_t ws_size,
                              hipStream_t stream) {
  (void)in_sizes; (void)n_in; (void)out_size; (void)ws_size;
  const float* w = (const float*)d_in[0];
  float* wp = (float*)d_ws;                                        // 8 * 962560 f32 (~30.8 MB)
  int* idxbuf = (int*)((char*)d_ws +
                       (size_t)BATCH * WP_STRIDE * sizeof(float)); // 8 * 1879 i32

  const long n1 = (long)BATCH * WP_STRIDE;
  build_wp<<<(int)((n1 + 255) / 256), 256, 0, stream>>>(w, wp);

  dim3 g2((NFRAMES + 3) / 4, BATCH);
  nccf_idx<<<g2, 128, 0, stream>>>(wp, idxbuf);

  const int n3 = BATCH * OUT_FRAMES;
  median_out<<<(n3 + 255) / 256, 256, 0, stream>>>(idxbuf, (float*)d_out);
}